// RepulsionXTB_56143812494147
// MI455X (gfx1250) — compile-verified
//
#include <hip/hip_runtime.h>
#include <hip/hip_bf16.h>

// ---------------------------------------------------------------------------
// GFN1-xTB repulsion energy, MI455X (gfx1250).
// Memory-bound streaming kernel:
//   * 3-deep rotating global->LDS async prefetch (ASYNCcnt path) to keep
//     ~24 MB in flight across the device (23.3 TB/s x ~1us BW-delay product)
//   * per-block LDS f32 histogram (ds_add_f32) + one global atomic flush
//   * fused single-exp math: rep = (y/d) * exp(ecut - alpha*d^k), d^1.5 = d*sqrt(d)
// ---------------------------------------------------------------------------

#define NTHREADS 256
#define TILE     1024            // pairs per tile = NTHREADS * 4 (one b128/lane/stream)
#define DEPTH    3               // async pipeline depth (rotating buffers)
#define MAX_LDS_BINS 4096

static __device__ __forceinline__ void async_load_b128(unsigned lds_addr,
                                                       const void* gptr) {
  // GLOBAL_LOAD_ASYNC_TO_LDS_B128 (GV mode): per-lane 16B global -> LDS.
  // Tracked with ASYNCcnt; async loads complete in order.
  asm volatile("global_load_async_to_lds_b128 %0, %1, off"
               :
               : "v"(lds_addr), "v"(gptr)
               : "memory");
}

#define WAIT_ASYNCCNT_6() asm volatile("s_wait_asynccnt 6" ::: "memory")
#define WAIT_ASYNCCNT_3() asm volatile("s_wait_asynccnt 3" ::: "memory")
#define WAIT_ASYNCCNT_0() asm volatile("s_wait_asynccnt 0" ::: "memory")

__global__ void RepulsionXTB_zero_kernel(float* __restrict__ out, int n) {
  int i = blockIdx.x * blockDim.x + threadIdx.x;
  if (i < n) out[i] = 0.0f;
}

__device__ __forceinline__ float pair_energy(float dist_ang, int n0, int n1,
                                             const int* __restrict__ elem,
                                             const float4* __restrict__ tbl,
                                             float rc, float inv_rc) {
  const float A2B = 1.8897261258369282f;
  float d = fmaxf(dist_ang, 1e-7f) * A2B;            // bohr
  int s0 = elem[n0];
  int s1 = elem[n1];
  float4 t = tbl[(s0 << 2) | s1];                    // {y, sqrt_alpha, k_rep}
  // k_rep is 1.0 (H-H) or 1.5 in GFN1-xTB: d^k = d or d*sqrt(d)
  float p = (t.z == 1.0f) ? d : d * __builtin_sqrtf(d);
  // smooth cutoff exponent: 1 - 1/max(1-(d/rc)^2, 1e-10), fused into one exp
  float x  = d * inv_rc;
  float om = fmaxf(1.0f - x * x, 1e-10f);
  float ecut = 1.0f - __builtin_amdgcn_rcpf(om);
  float arg  = ecut - t.y * p;
  float rep  = t.x * __builtin_amdgcn_rcpf(d) * __expf(arg);
  return (d < rc) ? rep : 0.0f;
}

__global__ __launch_bounds__(NTHREADS) void RepulsionXTB_main_kernel(
    const int*   __restrict__ elem,   // [M*A] species 0..3
    const int*   __restrict__ nbr,    // [2*P] flat: row0 then row1
    const float* __restrict__ dist,   // [P] angstrom
    const float* __restrict__ y_ab,   // [16]
    const float* __restrict__ sa_ab,  // [16]
    const float* __restrict__ kr_ab,  // [16]
    float*       __restrict__ out,    // [M]
    int M, int P, int shift, int A, int ntiles, int tiles_per_block) {
  const float A2B = 1.8897261258369282f;
  const float rc = 5.2f * A2B;
  const float inv_rc = 1.0f / (5.2f * A2B);

  __shared__ float4 tbl[16];                    // 256 B combined param table
  __shared__ float  acc[MAX_LDS_BINS];          // 16 KB per-block histogram
  __shared__ float  s_d [DEPTH][TILE];          // 12 KB rotating dist buffers
  __shared__ int    s_n0[DEPTH][TILE];          // 12 KB rotating nbr0 buffers
  __shared__ int    s_n1[DEPTH][TILE];          // 12 KB rotating nbr1 buffers

  int tid = threadIdx.x;
  if (tid < 16) tbl[tid] = make_float4(y_ab[tid], sa_ab[tid], kr_ab[tid], 0.f);
  const bool lds_bins = (M <= MAX_LDS_BINS);
  if (lds_bins)
    for (int i = tid; i < M; i += NTHREADS) acc[i] = 0.0f;
  __syncthreads();

  // This thread's LDS byte addresses (buffer 0); buffer b is +b*TILE*4 bytes.
  unsigned lad_d  = (unsigned)(unsigned long long)(&s_d [0][0]) + (unsigned)tid * 16u;
  unsigned lad_n0 = (unsigned)(unsigned long long)(&s_n0[0][0]) + (unsigned)tid * 16u;
  unsigned lad_n1 = (unsigned)(unsigned long long)(&s_n1[0][0]) + (unsigned)tid * 16u;

  long t0 = (long)blockIdx.x * tiles_per_block;
  long t1 = t0 + tiles_per_block;
  if (t1 > ntiles) t1 = ntiles;
  int n_my = (t1 > t0) ? (int)(t1 - t0) : 0;

  if (n_my > 0) {
    // Prologue: issue DEPTH-1 tiles ahead (3 b128s per lane each).
    int pre = (n_my < DEPTH - 1) ? n_my : (DEPTH - 1);
    for (int k = 0; k < pre; ++k) {
      size_t base = (size_t)(t0 + k) * TILE + (size_t)tid * 4;
      unsigned off = (unsigned)k * (TILE * 4u);
      async_load_b128(lad_d  + off, dist + base);
      async_load_b128(lad_n0 + off, nbr + base);
      async_load_b128(lad_n1 + off, nbr + (size_t)P + base);
    }
    int cur = 0;
    for (int c = 0; c < n_my; ++c) {
      if (c + DEPTH - 1 < n_my) {
        // Issue tile c+2 into the buffer freed in iteration c-1 (its ds
        // reads were waited on before use there -> no WAR hazard).
        int ib = cur + (DEPTH - 1); if (ib >= DEPTH) ib -= DEPTH;
        unsigned off = (unsigned)ib * (TILE * 4u);
        size_t base = (size_t)(t0 + c + DEPTH - 1) * TILE + (size_t)tid * 4;
        async_load_b128(lad_d  + off, dist + base);
        async_load_b128(lad_n0 + off, nbr + base);
        async_load_b128(lad_n1 + off, nbr + (size_t)P + base);
      }
      // In-order completion: leave 3 per pending newer tile outstanding.
      int pending = n_my - 1 - c;
      if (pending >= 2)      WAIT_ASYNCCNT_6();
      else if (pending == 1) WAIT_ASYNCCNT_3();
      else                   WAIT_ASYNCCNT_0();

      // Each thread consumes exactly the 4 pairs it async-loaded.
      float4 dv  = *(const float4*)&s_d [cur][tid * 4];
      int4   n0v = *(const int4*)  &s_n0[cur][tid * 4];
      int4   n1v = *(const int4*)  &s_n1[cur][tid * 4];

      float r0 = pair_energy(dv.x, n0v.x, n1v.x, elem, tbl, rc, inv_rc);
      float r1 = pair_energy(dv.y, n0v.y, n1v.y, elem, tbl, rc, inv_rc);
      float r2 = pair_energy(dv.z, n0v.z, n1v.z, elem, tbl, rc, inv_rc);
      float r3 = pair_energy(dv.w, n0v.w, n1v.w, elem, tbl, rc, inv_rc);

      int m0 = (shift >= 0) ? ((unsigned)n0v.x >> shift) : (n0v.x / A);
      int m1 = (shift >= 0) ? ((unsigned)n0v.y >> shift) : (n0v.y / A);
      int m2 = (shift >= 0) ? ((unsigned)n0v.z >> shift) : (n0v.z / A);
      int m3 = (shift >= 0) ? ((unsigned)n0v.w >> shift) : (n0v.w / A);
      if (lds_bins) {
        // ds_add_f32 (no return); skip lanes beyond the cutoff (~14% of pairs)
        if (r0 != 0.0f) atomicAdd(&acc[m0], r0);
        if (r1 != 0.0f) atomicAdd(&acc[m1], r1);
        if (r2 != 0.0f) atomicAdd(&acc[m2], r2);
        if (r3 != 0.0f) atomicAdd(&acc[m3], r3);
      } else {
        if (r0 != 0.0f) atomicAdd(&out[m0], r0);
        if (r1 != 0.0f) atomicAdd(&out[m1], r1);
        if (r2 != 0.0f) atomicAdd(&out[m2], r2);
        if (r3 != 0.0f) atomicAdd(&out[m3], r3);
      }
      cur = (cur + 1 == DEPTH) ? 0 : cur + 1;
    }
  }

  // Remainder pairs (P not a multiple of TILE, or misaligned fallback):
  // handled scalar by block 0.
  if (blockIdx.x == 0) {
    for (long i = (long)ntiles * TILE + tid; i < (long)P; i += NTHREADS) {
      float dd = dist[i];
      int n0 = nbr[i];
      int n1 = nbr[(size_t)P + i];
      float r = pair_energy(dd, n0, n1, elem, tbl, rc, inv_rc);
      int m = (shift >= 0) ? ((unsigned)n0 >> shift) : (n0 / A);
      if (r != 0.0f) {
        if (lds_bins) atomicAdd(&acc[m], r);
        else          atomicAdd(&out[m], r);
      }
    }
  }

  __syncthreads();
  if (lds_bins && (n_my > 0 || blockIdx.x == 0)) {
    for (int i = tid; i < M; i += NTHREADS) {
      float v = acc[i];
      if (v != 0.0f) atomicAdd(&out[i], v);   // global_atomic_add_f32
    }
  }
}

extern "C" void kernel_launch(void* const* d_in, const int* in_sizes, int n_in,
                              void* d_out, int out_size, void* d_ws, size_t ws_size,
                              hipStream_t stream) {
  const int*   elem = (const int*)d_in[0];    // element_idxs [M*A]
  const int*   nbr  = (const int*)d_in[1];    // neighbor_idxs [2,P] flat
  const float* dist = (const float*)d_in[2];  // distances [P]
  const float* y    = (const float*)d_in[3];  // y_ab [16]
  const float* sa   = (const float*)d_in[4];  // sqrt_alpha_ab [16]
  const float* kr   = (const float*)d_in[5];  // k_rep_ab [16]
  float* out = (float*)d_out;

  int M = out_size;                           // 2048
  int P = in_sizes[2];                        // 8388608
  int A = (M > 0) ? in_sizes[0] / M : 1;      // 64
  int shift = -1;
  if (A > 0 && (A & (A - 1)) == 0) {
    shift = 0;
    while ((1 << shift) != A) shift++;
  }

  // Async b128 path requires 16B-aligned stream bases: row1 of nbr starts at
  // element P, so P must be a multiple of 4. Otherwise everything goes
  // through the scalar tail.
  int ntiles = ((P & 3) == 0) ? (P / TILE) : 0;
  int grid = ntiles > 0 ? (ntiles < 1024 ? ntiles : 1024) : 1;
  int tiles_per_block = (ntiles + grid - 1) / grid;
  if (tiles_per_block < 1) tiles_per_block = 1;

  RepulsionXTB_zero_kernel<<<(M + 255) / 256, 256, 0, stream>>>(out, M);
  RepulsionXTB_main_kernel<<<grid, NTHREADS, 0, stream>>>(
      elem, nbr, dist, y, sa, kr, out, M, P, shift, A, ntiles, tiles_per_block);
}